// AttentionLayer_4088808865865
// MI455X (gfx1250) — compile-verified
//
#include <hip/hip_runtime.h>
#include <cstdint>
#include <cstddef>

// ---------------------------------------------------------------------------
// MI455X (gfx1250) fused multi-head attention, fp32 end-to-end.
//   HEAD_DIM = 16  -> exactly one WMMA tile per head slice.
//   All matmuls use V_WMMA_F32_16X16X4_F32 (wave32, D = A(16x4) x B(4x16) + C).
// Memory-bound problem (~350MB traffic @ 23.3TB/s => ~15us floor); design:
//   * fuse score -> online softmax -> attn@V (attn never touches HBM)
//   * coalesce the dominant 268MB score write via LDS [n][m][h] staging
//   * attn_mask tiles double-buffered with GLOBAL_LOAD_ASYNC_TO_LDS_B32
//     (ASYNCcnt) so the fetch overlaps the WMMA + softmax work of the
//     current tile; loaded once per workgroup (8 heads share).
// ---------------------------------------------------------------------------

typedef float v2f __attribute__((ext_vector_type(2)));
typedef float v8f __attribute__((ext_vector_type(8)));

#define BSZ   8
#define NMAX  1024
#define HID   128
#define NHEAD 8
#define FDIM  16      // head_dim

// workspace layout (float offsets): q,k,v per-head [B][H][N][16]; o [B][N][128]
#define Q_OFF 0u
#define K_OFF 1048576u
#define V_OFF 2097152u
#define O_OFF 3145728u

// D = A(16x4) * B(4x16) + C   (fp32 WMMA)
// A layout: lane L -> row M=L%16; VGPRs hold K = (L/16)*2 + {0,1}
// B layout: lane L -> col N=L%16; VGPRs hold K = (L/16)*2 + {0,1}
// C/D layout: lane L -> col N=L%16; VGPR r -> row M = r + 8*(L/16)
__device__ __forceinline__ v8f wmma_f32(v2f a, v2f b, v8f c) {
  return __builtin_amdgcn_wmma_f32_16x16x4_f32(false, a, false, b, (short)0, c,
                                               false, false);
}

// Async memory->LDS copy, 4 bytes per lane, GVS addressing:
//   copies from global address sbase plus voff to the LDS address in lds_off   (tracked by ASYNCcnt)
__device__ __forceinline__ void async_load_b32(unsigned lds_off, unsigned voff,
                                               const float* sbase) {
  asm volatile("global_load_async_to_lds_b32 %0, %1, %2 offset:0"
               :
               : "v"(lds_off), "v"(voff), "s"(sbase)
               : "memory");
}
__device__ __forceinline__ void wait_async0() {
  asm volatile("s_wait_asynccnt 0x0" ::: "memory");
}

// ===========================================================================
// Kernel 1: Q/K/V projections.  One wave = one 16x16 tile of one head of one
// of q/k/v.  y = x @ W + bias, columns picked per head (hidden = f*8 + h),
// stored per-head [B][H][N][16].  1/sqrt(16) folded into Q.
// grid.x = 3 * B * (N/16) * H = 12288, block = 32
// ===========================================================================
__global__ void __launch_bounds__(32)
qkv_proj(const float* __restrict__ x,
         const float* __restrict__ Wq, const float* __restrict__ bq,
         const float* __restrict__ Wk, const float* __restrict__ bk,
         const float* __restrict__ Wv, const float* __restrict__ bv,
         float* __restrict__ ws) {
  const int lane = threadIdx.x;
  const int l16  = lane & 15;
  const int half = lane >> 4;

  int idx = blockIdx.x;
  const int h     = idx & 7;  idx >>= 3;
  const int nt    = idx & 63; idx >>= 6;
  const int b     = idx & 7;
  const int which = idx >> 3;                      // 0=q 1=k 2=v

  const float* W    = (which == 0) ? Wq : (which == 1) ? Wk : Wv;
  const float* bias = (which == 0) ? bq : (which == 1) ? bk : bv;
  float* dst = ws + ((which == 0) ? Q_OFF : (which == 1) ? K_OFF : V_OFF)
                  + ((size_t)(b * NHEAD + h) * NMAX + (size_t)nt * 16) * FDIM;
  const float* X = x + ((size_t)b * NMAX + (size_t)nt * 16) * HID;

  v8f acc = {};
  for (int k0 = 0; k0 < HID; k0 += 4) {
    const int ka = k0 + half * 2;
    v2f a, bb;
    a.x  = X[l16 * HID + ka];
    a.y  = X[l16 * HID + ka + 1];
    bb.x = W[ka * HID + l16 * NHEAD + h];          // W[k][f*8+h], f = l16
    bb.y = W[(ka + 1) * HID + l16 * NHEAD + h];
    acc = wmma_f32(a, bb, acc);
  }

  const float bcol = bias[l16 * NHEAD + h];
  const float sc   = (which == 0) ? 0.25f : 1.0f;  // fold 1/sqrt(head_dim) into Q
  for (int r = 0; r < 8; ++r) {
    const int row = r + 8 * half;
    dst[row * FDIM + l16] = (acc[r] + bcol) * sc;
  }
}

// ===========================================================================
// Kernel 2: fused  score = K.Q^T + mask - pad  (written to d_out),
//                  online softmax over m,  O = attn @ V.
// Workgroup = 8 waves = the 8 heads of one (b, 16-row n-tile).
// grid.x = B * (N/16) = 512, block = 256
// ===========================================================================
__global__ void __launch_bounds__(256)
attn_kernel(const float* __restrict__ mask,
            const unsigned char* __restrict__ key_pad,
            const float* __restrict__ ws,
            float* __restrict__ score_out,
            float* __restrict__ o_ws) {
  __shared__ __align__(16) float sbuf[16 * 16 * NHEAD]; // [n][m][h] score tile
  __shared__ float mbuf[2][16][16];                     // mask tile, double buffer
  __shared__ float pbuf[NHEAD][16][16];                 // per-wave P (layout bridge)

  const int t    = threadIdx.x;
  const int lane = t & 31;
  const int h    = t >> 5;                              // wave id == head
  const int l16  = lane & 15;
  const int half = lane >> 4;

  const int b  = blockIdx.x >> 6;
  const int n0 = (blockIdx.x & 63) * 16;

  const float* Qh = ws + Q_OFF + (size_t)(b * NHEAD + h) * NMAX * FDIM;
  const float* Kh = ws + K_OFF + (size_t)(b * NHEAD + h) * NMAX * FDIM;
  const float* Vh = ws + V_OFF + (size_t)(b * NHEAD + h) * NMAX * FDIM;

  // mask batch slice base (SGPR pair) + this thread's element inside a tile
  const float* mb = mask + ((size_t)b << 20);
  const int mrow = t >> 4, mcol = t & 15;               // element covered by t
  const unsigned mrow_goff = (unsigned)(n0 + mrow) * 1024u * 4u;
  // LDS byte addresses of this thread's slot in each mask buffer
  const unsigned mslot0 = (unsigned)(uintptr_t)&mbuf[0][mrow][mcol];
  const unsigned mslot1 = (unsigned)(uintptr_t)&mbuf[1][mrow][mcol];

  v8f oacc = {};
  float rm[8], rl[8];                                   // row of oacc[r] = r + 8*half
  for (int r = 0; r < 8; ++r) { rm[r] = -3.0e38f; rl[r] = 0.0f; }

  // prologue: async-fetch mask tile 0 into buffer 0
  async_load_b32(mslot0, mrow_goff + (unsigned)mcol * 4u, mb);

  for (int mt = 0; mt < 64; ++mt) {
    const int m0  = mt * 16;
    const int cur = mt & 1;

    // ---- S[n][m] = sum_f K[n0+n][f] * Q[m0+m][f]  (scale pre-folded into Q)
    v8f acc = {};
    for (int kk = 0; kk < 4; ++kk) {
      const int f = kk * 4 + half * 2;
      v2f a, bb;
      a.x  = Kh[(n0 + l16) * FDIM + f];
      a.y  = Kh[(n0 + l16) * FDIM + f + 1];
      bb.x = Qh[(m0 + l16) * FDIM + f];               // B[f][m] = Q[m][f]
      bb.y = Qh[(m0 + l16) * FDIM + f + 1];
      acc = wmma_f32(a, bb, acc);
    }

    wait_async0();      // this wave's slice of mbuf[cur] has landed in LDS
    __syncthreads();    // all waves' slices landed; prev sbuf consumers done

    // prefetch next mask tile into the other buffer (overlaps epilogue+WMMA)
    if (mt < 63) {
      async_load_b32(cur ? mslot0 : mslot1,
                     mrow_goff + (unsigned)(m0 + 16 + mcol) * 4u, mb);
    }

    const float kp = key_pad[b * NMAX + m0 + l16] ? 1.0e10f : 0.0f;

    // ---- epilogue: mask/pad, stage score, online-softmax stats, P tile
    for (int r = 0; r < 8; ++r) {
      const int row = r + 8 * half;
      float s = acc[r] + mbuf[cur][row][l16] - kp;
      sbuf[row * 128 + l16 * 8 + h] = s;              // [n][m][h] staging

      float tmax = s;                                  // row-max over 16 lanes
      for (int off = 8; off; off >>= 1)
        tmax = fmaxf(tmax, __shfl_xor(tmax, off, 16));
      const float nm   = fmaxf(rm[r], tmax);
      const float resc = __expf(rm[r] - nm);
      const float p    = __expf(s - nm);
      float tsum = p;                                  // row-sum over 16 lanes
      for (int off = 8; off; off >>= 1)
        tsum += __shfl_xor(tsum, off, 16);
      rl[r]   = rl[r] * resc + tsum;
      rm[r]   = nm;
      oacc[r] = oacc[r] * resc;
      pbuf[h][row][l16] = p;                           // private to this wave
    }

    __syncthreads();  // sbuf complete across all 8 heads

    // ---- coalesced score write: thread t covers sbuf[t*8 .. t*8+7], which is
    // row n = t/16, cols (m,h) = (t%16, 0..7); rows are 512B contiguous runs.
    {
      const float4* src = (const float4*)&sbuf[t * 8];
      float* drow = score_out + ((size_t)(b * NMAX + n0 + (t >> 4)) << 13)
                              + (size_t)m0 * 8 + (t & 15) * 8;
      ((float4*)drow)[0] = src[0];
      ((float4*)drow)[1] = src[1];
    }

    // ---- O += P(16n x 16m) @ V(16m x 16f); P re-read from LDS in A layout
    for (int kk = 0; kk < 4; ++kk) {
      const int k = kk * 4 + half * 2;
      v2f a, bb;
      a.x  = pbuf[h][l16][k];
      a.y  = pbuf[h][l16][k + 1];
      bb.x = Vh[(m0 + k) * FDIM + l16];
      bb.y = Vh[(m0 + k + 1) * FDIM + l16];
      oacc = wmma_f32(a, bb, oacc);
    }
  }

  // ---- normalize and write O into [B][N][128] with hidden = f*8 + h
  for (int r = 0; r < 8; ++r) {
    const int row = r + 8 * half;
    const float inv = 1.0f / rl[r];
    o_ws[(size_t)(b * NMAX + n0 + row) * HID + l16 * NHEAD + h] = oacc[r] * inv;
  }
}

// ===========================================================================
// Kernel 3: out = o @ Wo + bo.  One wave = one 16x16 output tile.
// grid.x = (B*N/16) * (128/16) = 4096, block = 32
// ===========================================================================
__global__ void __launch_bounds__(32)
out_proj(const float* __restrict__ o, const float* __restrict__ Wo,
         const float* __restrict__ bo, float* __restrict__ out) {
  const int lane = threadIdx.x;
  const int l16  = lane & 15;
  const int half = lane >> 4;
  const int j0   = (blockIdx.x & 7) * 16;
  const int rt   = blockIdx.x >> 3;

  const float* O = o + (size_t)rt * 16 * HID;
  v8f acc = {};
  for (int k0 = 0; k0 < HID; k0 += 4) {
    const int ka = k0 + half * 2;
    v2f a, bb;
    a.x  = O[l16 * HID + ka];
    a.y  = O[l16 * HID + ka + 1];
    bb.x = Wo[ka * HID + j0 + l16];
    bb.y = Wo[(ka + 1) * HID + j0 + l16];
    acc = wmma_f32(a, bb, acc);
  }
  const float bcol = bo[j0 + l16];
  for (int r = 0; r < 8; ++r) {
    const int row = r + 8 * half;
    out[((size_t)rt * 16 + row) * HID + j0 + l16] = acc[r] + bcol;
  }
}

// ===========================================================================
extern "C" void kernel_launch(void* const* d_in, const int* in_sizes, int n_in,
                              void* d_out, int out_size, void* d_ws,
                              size_t ws_size, hipStream_t stream) {
  (void)in_sizes; (void)n_in; (void)out_size; (void)ws_size;
  const float*         x       = (const float*)d_in[0];
  const unsigned char* key_pad = (const unsigned char*)d_in[1]; // jax bool = 1 byte
  const float*         mask    = (const float*)d_in[2];
  const float*         Wq      = (const float*)d_in[3];
  const float*         bq      = (const float*)d_in[4];
  const float*         Wk      = (const float*)d_in[5];
  const float*         bk      = (const float*)d_in[6];
  const float*         Wv      = (const float*)d_in[7];
  const float*         bv      = (const float*)d_in[8];
  const float*         Wo      = (const float*)d_in[9];
  const float*         bo      = (const float*)d_in[10];

  float* out   = (float*)d_out;                           // [8,1024,128]
  float* score = out + (size_t)BSZ * NMAX * HID;          // [8,1024,1024,8]
  float* ws    = (float*)d_ws;                            // needs 16 MB

  qkv_proj<<<3 * BSZ * (NMAX / 16) * NHEAD, 32, 0, stream>>>(
      x, Wq, bq, Wk, bk, Wv, bv, ws);
  attn_kernel<<<BSZ * (NMAX / 16), 256, 0, stream>>>(
      mask, key_pad, ws, score, ws + O_OFF);
  out_proj<<<(BSZ * NMAX / 16) * (HID / 16), 32, 0, stream>>>(
      ws + O_OFF, Wo, bo, out);
}